// Aggregation_22282290332092
// MI455X (gfx1250) — compile-verified
//
#include <hip/hip_runtime.h>
#include <hip/hip_bf16.h>
#include <math.h>

// ---------------------------------------------------------------------------
// MI455X (gfx1250) fused single-query multi-head attention aggregation.
//
// Shapes (fixed by setup_inputs): D=1024, H=16, hd=64, n_sents=32, n_ents=32,
// B=64, SE=1024.  All fp32.
//
// Algebra: K-proj folded into query (qk = q@Wk), V-proj deferred past softmax
// via weighted entities we[h,:] = sum_e attn[h,e]*ent[e,:]  (softmax sums to 1).
// Pipeline:
//   prep_q    : q = (query@Wq^T + bq) * 1/sqrt(64)
//   prep_qk   : qk[h,d] = sum_hd q[h,hd]*Wk[h*64+hd,d];  sbias[h]=q_h.bk_h
//   attn_we   : per (s,b): scores=ent@qk^T (WMMA), softmax, we=attn@ent (WMMA)
//   ctx       : per (s,h): ctx[b,hd] = we @ Wv_h^T + bv  (WMMA GEMM 64x64xK1024)
//   out       : result = ctx @ Wo^T + bo, zero all-masked rows (WMMA GEMM)
//
// Data movement: entities read once (256 MB), we round-trip (256 MB), ctx +
// weights (~32 MB) => ~0.55 GB @ 23.3 TB/s ~= 23 us; compute ~13 GFLOP fp32
// WMMA, far below the matrix-op ceiling => memory-bound, so staging uses the
// CDNA5 async global->LDS path (ASYNCcnt) when the toolchain exposes it.
//
// Workspace (floats): q[1024] @0, sbias[16] @1024, allmask[2048] @2048,
// qk[16384] @8192, we[33554432] @32768, ctx[2097152] after we.  ~143 MB.
// ---------------------------------------------------------------------------

typedef __attribute__((ext_vector_type(2))) float v2f;
typedef __attribute__((ext_vector_type(8))) float v8f;
typedef int v4i_ __attribute__((vector_size(16)));  // matches builtin proto

__device__ __forceinline__ v8f wmma4(v2f a, v2f b, v8f c) {
  // D(16x16,f32) = A(16x4,f32) * B(4x16,f32) + C
  return __builtin_amdgcn_wmma_f32_16x16x4_f32(false, a, false, b, (short)0, c,
                                               false, false);
}

#if defined(__gfx1250__) && \
    __has_builtin(__builtin_amdgcn_global_load_async_to_lds_b128)
#define ASYNC_LDS 1
#else
#define ASYNC_LDS 0
#endif

// 16-byte global -> LDS copy. CDNA5: GLOBAL_LOAD_ASYNC_TO_LDS_B128 (no VGPR
// landing zone, tracked by ASYNCcnt). Fallback: through-VGPR float4 copy.
__device__ __forceinline__ void stage16(float* lds, const float* g) {
#if ASYNC_LDS
  __builtin_amdgcn_global_load_async_to_lds_b128(
      (v4i_*)g, (__attribute__((address_space(3))) v4i_*)lds, 0, 0);
#else
  *(float4*)lds = *(const float4*)g;
#endif
}

__device__ __forceinline__ void stage_wait() {
#if ASYNC_LDS
#if __has_builtin(__builtin_amdgcn_s_wait_asynccnt)
  __builtin_amdgcn_s_wait_asynccnt(0);
#else
  asm volatile("s_wait_asynccnt 0" ::: "memory");
#endif
#endif
}

#define D 1024
#define H 16
#define NS 32
#define NE 32
#define BB 64

// ------------------------------ prep kernels -------------------------------

__global__ __launch_bounds__(256) void prep_q(const float* __restrict__ query,
                                              const float* __restrict__ in_w,
                                              const float* __restrict__ in_b,
                                              float* __restrict__ q) {
  int j = blockIdx.x * 256 + threadIdx.x;
  if (j < D) {
    float acc = in_b[j];
    const float* wr = in_w + (size_t)j * D;
    for (int i = 0; i < D; ++i) acc += query[i] * wr[i];
    q[j] = acc * 0.125f;  // 1/sqrt(hd=64)
  }
}

__global__ __launch_bounds__(256) void prep_qk(const float* __restrict__ q,
                                               const float* __restrict__ in_w,
                                               const float* __restrict__ in_b,
                                               float* __restrict__ qk,
                                               float* __restrict__ sbias) {
  // grid 64: block = h*4 + dq ; thread covers one output column d
  int h = blockIdx.x >> 2, dq = blockIdx.x & 3;
  int d = dq * 256 + threadIdx.x;
  const float* Wk = in_w + (size_t)D * D;  // rows D..2D-1
  float acc = 0.f;
  for (int t = 0; t < 64; ++t)
    acc += q[h * 64 + t] * Wk[(size_t)(h * 64 + t) * D + d];
  qk[h * D + d] = acc;
  if (dq == 0 && threadIdx.x == 0) {
    const float* bk = in_b + D;
    float sb = 0.f;
    for (int t = 0; t < 64; ++t) sb += q[h * 64 + t] * bk[h * 64 + t];
    sbias[h] = sb;
  }
}

// --------------------------- attention + we pass ---------------------------

#define ESTR 1028  // 1024+4: bank = 4*row + k (mod 64) -> conflict-free, 8B ok

__global__ __launch_bounds__(256) void attn_we_kernel(
    const float* __restrict__ entities, const unsigned char* __restrict__ pm,
    const float* __restrict__ qk, const float* __restrict__ sbias,
    float* __restrict__ we, float* __restrict__ allmask_ws) {
  __shared__ float entL[NE * ESTR];    // 32 x 1024 slab      (~128.5 KB)
  __shared__ float scoresL[H * 36];    // 16 x 32 (+pad)
  __shared__ float attnL[H * 36];
  __shared__ float red[8 * 256];       // partial 16x16 tiles (8 KB)

  const int s = blockIdx.x >> 6;
  const int b = blockIdx.x & 63;
  const int tid = threadIdx.x;
  const int lane = tid & 31, wave = tid >> 5;
  const int r16 = lane & 15, hi = lane >> 4;

  // stage entity slab: ent[e][d] = entities[(s*32+e)*B*D + b*D + d]
  const float* eb = entities + (size_t)s * NE * BB * D + (size_t)b * D;
  for (int i = 0; i < 32; ++i) {
    int idx = tid + i * 256;            // 8192 float4 total
    int row = idx >> 8;                 // 256 float4 per row
    int c4 = (idx & 255) << 2;
    stage16(&entL[row * ESTR + c4], &eb[(size_t)row * BB * D + c4]);
  }
  stage_wait();
  __syncthreads();

  // ---- scores[h,e] = qk[h,:] . ent[e,:]  (WMMA, waves = etile x Kquarter) --
  {
    const int et = wave & 1, kq = wave >> 1;
    v8f c = {};
    const float* br = &entL[(et * 16 + r16) * ESTR + 2 * hi];
    const float* ar = &qk[r16 * D + 2 * hi];  // qk hot in WGP$/L2
    for (int k = kq * 256; k < kq * 256 + 256; k += 4) {
      v2f a = *(const v2f*)(ar + k);
      v2f bb = *(const v2f*)(br + k);
      c = wmma4(a, bb, c);
    }
    float* rp = &red[wave * 256];
#pragma unroll
    for (int i = 0; i < 8; ++i) rp[i * 32 + lane] = c[i];
  }
  __syncthreads();
  if (wave < 2) {  // reduce 4 K-partials per e-tile, emit scores[h][e]
    const int et = wave;
#pragma unroll
    for (int i = 0; i < 8; ++i) {
      float v = red[(0 + et) * 256 + i * 32 + lane] +
                red[(2 + et) * 256 + i * 32 + lane] +
                red[(4 + et) * 256 + i * 32 + lane] +
                red[(6 + et) * 256 + i * 32 + lane];
      int h = i + 8 * hi;         // C layout: row = vgpr + 8*(lane>=16)
      int e = et * 16 + r16;      // col = lane%16
      scoresL[h * 36 + e] = v;
    }
  }
  __syncthreads();

  // ---- softmax over e per head (16 threads), padding mask, allmask flag ----
  if (tid == 0) {
    bool allm = true;
    for (int e = 0; e < NE; ++e)
      allm = allm && (pm[((size_t)s * NE + e) * BB + b] != 0);
    allmask_ws[s * BB + b] = allm ? 1.f : 0.f;
  }
  if (tid < H) {
    const int h = tid;
    const float sb = sbias[h];
    float sc[NE];
    float mx = -3.4e38f;
#pragma unroll
    for (int e = 0; e < NE; ++e) {
      float v = scoresL[h * 36 + e] + sb;
      if (pm[((size_t)s * NE + e) * BB + b]) v = -3.4e38f;
      sc[e] = v;
      mx = fmaxf(mx, v);
    }
    float sum = 0.f;
#pragma unroll
    for (int e = 0; e < NE; ++e) {
      float ex = (sc[e] <= -3.0e38f) ? 0.f : __expf(sc[e] - mx);
      sc[e] = ex;
      sum += ex;
    }
    float inv = (sum > 0.f) ? 1.f / sum : 0.f;
#pragma unroll
    for (int e = 0; e < NE; ++e) attnL[h * 36 + e] = sc[e] * inv;
  }
  __syncthreads();

  // ---- we[h,d] = attn(16x32) @ ent(32x1024)  (WMMA, 8 d-tiles per wave) ----
  float* web = we + ((size_t)(s * H) * BB + b) * D;  // row h at +h*BB*D
  for (int j = 0; j < 8; ++j) {
    const int n0 = (wave * 8 + j) * 16;
    v8f c = {};
#pragma unroll
    for (int k = 0; k < NE; k += 4) {
      int kk = k + 2 * hi;
      v2f a = *(const v2f*)&attnL[r16 * 36 + kk];
      v2f bb;
      bb.x = entL[kk * ESTR + n0 + r16];
      bb.y = entL[(kk + 1) * ESTR + n0 + r16];
      c = wmma4(a, bb, c);
    }
#pragma unroll
    for (int i = 0; i < 8; ++i) {
      int h = i + 8 * hi;
      web[(size_t)h * BB * D + n0 + r16] = c[i];
    }
  }
}

// ------------------- shared 64x64-tile WMMA GEMM (C=A.B^T) ------------------

#define GSTR 68  // 64+4: conflict-free fragment reads, 8B/16B aligned

__device__ __forceinline__ void gemm_tile64(
    const float* __restrict__ A, int lda,        // 64 x K
    const float* __restrict__ B, int ldb,        // 64 x K (weights [N,K])
    const float* __restrict__ bias,              // per-N or nullptr
    const float* __restrict__ rowmask,           // per-M zero flag or nullptr
    float* __restrict__ C, int ldc, int K, float* As, float* Bs) {
  const int tid = threadIdx.x;
  const int lane = tid & 31, wave = tid >> 5;
  const int wr = wave >> 1, wc = wave & 1;  // wave -> 16x32 C region
  const int r16 = lane & 15, hi = lane >> 4;
  v8f c0 = {}, c1 = {};
  for (int k0 = 0; k0 < K; k0 += 64) {
#pragma unroll
    for (int i = 0; i < 4; ++i) {
      int idx = tid + i * 256;  // 1024 float4 per matrix
      int row = idx >> 4;
      int c4 = (idx & 15) << 2;
      stage16(&As[row * GSTR + c4], &A[(size_t)row * lda + k0 + c4]);
      stage16(&Bs[row * GSTR + c4], &B[(size_t)row * ldb + k0 + c4]);
    }
    stage_wait();
    __syncthreads();
    const float* ar = &As[(wr * 16 + r16) * GSTR + 2 * hi];
    const float* b0 = &Bs[(wc * 32 + r16) * GSTR + 2 * hi];
    const float* b1 = &Bs[(wc * 32 + 16 + r16) * GSTR + 2 * hi];
#pragma unroll
    for (int k = 0; k < 64; k += 4) {
      v2f a = *(const v2f*)(ar + k);
      v2f bb0 = *(const v2f*)(b0 + k);
      v2f bb1 = *(const v2f*)(b1 + k);
      c0 = wmma4(a, bb0, c0);
      c1 = wmma4(a, bb1, c1);
    }
    __syncthreads();
  }
  const int n0 = wc * 32 + r16;
#pragma unroll
  for (int i = 0; i < 8; ++i) {
    int r = wr * 16 + 8 * hi + i;
    float v0 = c0[i] + (bias ? bias[n0] : 0.f);
    float v1 = c1[i] + (bias ? bias[n0 + 16] : 0.f);
    if (rowmask && rowmask[r] != 0.f) { v0 = 0.f; v1 = 0.f; }
    C[(size_t)r * ldc + n0] = v0;
    C[(size_t)r * ldc + n0 + 16] = v1;
  }
}

// ---- ctx[s,b,h*64+hd] = we[(s,h)] (64b x 1024) @ Wv_h^T (64 x 1024) + bv ---

__global__ __launch_bounds__(256) void ctx_kernel(const float* __restrict__ we,
                                                  const float* __restrict__ in_w,
                                                  const float* __restrict__ in_b,
                                                  float* __restrict__ ctx) {
  __shared__ float As[64 * GSTR];
  __shared__ float Bs[64 * GSTR];
  const int s = blockIdx.x >> 4, h = blockIdx.x & 15;
  const float* A = we + (size_t)(s * H + h) * BB * D;
  const float* B = in_w + (size_t)2 * D * D + (size_t)(h * 64) * D;  // Wv_h
  const float* bias = in_b + 2 * D + h * 64;                         // bv_h
  float* C = ctx + (size_t)s * BB * D + h * 64;
  gemm_tile64(A, D, B, D, bias, nullptr, C, D, D, As, Bs);
}

// ---- result = ctx (2048 x 1024) @ Wo^T + bo, zero all-masked rows ----------

__global__ __launch_bounds__(256) void out_kernel(
    const float* __restrict__ ctx, const float* __restrict__ out_w,
    const float* __restrict__ out_b, const float* __restrict__ allm,
    float* __restrict__ out) {
  __shared__ float As[64 * GSTR];
  __shared__ float Bs[64 * GSTR];
  const int mt = blockIdx.x >> 4, nt = blockIdx.x & 15;  // 32 x 16 tiles
  gemm_tile64(ctx + (size_t)mt * 64 * D, D, out_w + (size_t)nt * 64 * D, D,
              out_b + nt * 64, allm + mt * 64,
              out + (size_t)mt * 64 * D + nt * 64, D, D, As, Bs);
}

// ------------------------------- launcher ----------------------------------

extern "C" void kernel_launch(void* const* d_in, const int* in_sizes, int n_in,
                              void* d_out, int out_size, void* d_ws,
                              size_t ws_size, hipStream_t stream) {
  (void)in_sizes; (void)n_in; (void)out_size; (void)ws_size;
  const float* entities = (const float*)d_in[0];
  const unsigned char* pmask = (const unsigned char*)d_in[1];  // bool bytes
  // d_in[2] = n_sents (scalar 32, baked into shapes)
  const float* query = (const float*)d_in[3];
  const float* in_w = (const float*)d_in[4];
  const float* in_b = (const float*)d_in[5];
  const float* out_w = (const float*)d_in[6];
  const float* out_b = (const float*)d_in[7];

  float* ws = (float*)d_ws;
  float* q = ws;                  // 1024
  float* sbias = ws + 1024;       // 16
  float* allm = ws + 2048;        // 2048
  float* qk = ws + 8192;          // 16*1024
  float* we = ws + 32768;         // 32*16*64*1024 = 33,554,432
  float* ctx = we + 33554432;     // 2048*1024
  float* out = (float*)d_out;

  prep_q<<<4, 256, 0, stream>>>(query, in_w, in_b, q);
  prep_qk<<<64, 256, 0, stream>>>(q, in_w, in_b, qk, sbias);
  attn_we_kernel<<<NS * BB, 256, 0, stream>>>(entities, pmask, qk, sbias, we, allm);
  ctx_kernel<<<NS * H, 256, 0, stream>>>(we, in_w, in_b, ctx);
  out_kernel<<<(2048 / 64) * (D / 64), 256, 0, stream>>>(ctx, out_w, out_b, allm, out);
}